// UnbalancedSinkhorn_55860344652248
// MI455X (gfx1250) — compile-verified
//
#include <hip/hip_runtime.h>

namespace {
constexpr int   N        = 8192;
constexpr int   D        = 64;
constexpr float EPS      = 0.05f;
constexpr float INV_EPS  = 20.0f;
constexpr float TAU      = 0.8f;
constexpr float LOGW     = -9.01091334728f;        // log(1/8192)
constexpr float POT_W    = EPS * LOGW;             // fa == gb
constexpr float RHO      = 0.2f;                   // EPS*TAU/(1-TAU)
constexpr int   ITERS    = 100;
constexpr int   RCH      = 32;                     // row chunks (256 rows each)
constexpr int   CHROWS   = N / RCH;                // 256
constexpr int   RES_ROWS = 5120;                   // 160 MB of S kept L2-resident (L2=192MB)
constexpr int   RES_CH   = RES_ROWS / CHROWS;      // 20 resident chunks
}

typedef __attribute__((ext_vector_type(2))) float v2f;
typedef __attribute__((ext_vector_type(4))) float v4f;
typedef __attribute__((ext_vector_type(8))) float v8f;

// ---- temporal-hint-aware loads ------------------------------------------
template <bool NT>
__device__ __forceinline__ v4f ld4(const v4f* p) {
  return NT ? __builtin_nontemporal_load(p) : *p;
}
template <bool NT>
__device__ __forceinline__ float ld1(const float* p) {
  return NT ? __builtin_nontemporal_load(p) : *p;
}

// ---- online logsumexp merge ---------------------------------------------
__device__ __forceinline__ void lse_merge(float& m, float& s, float m2, float s2) {
  float mn = fmaxf(m, m2);
  s = s * __expf(m - mn) + s2 * __expf(m2 - mn);
  m = mn;
}

// ---- setup: squared half-norms + zero potentials ------------------------
__global__ void setup_kernel(const float* __restrict__ X, const float* __restrict__ Y,
                             float* __restrict__ xn, float* __restrict__ yn,
                             float* __restrict__ f, float* __restrict__ g) {
  int i = blockIdx.x * 256 + threadIdx.x;          // 0..8191
  const v4f* xr = (const v4f*)(X + (size_t)i * D);
  const v4f* yr = (const v4f*)(Y + (size_t)i * D);
  float sx = 0.f, sy = 0.f;
#pragma unroll
  for (int k = 0; k < D / 4; ++k) {
    v4f a = xr[k];
    v4f b = yr[k];
    sx += a.x * a.x + a.y * a.y + a.z * a.z + a.w * a.w;
    sy += b.x * b.x + b.y * b.y + b.z * b.z + b.w * b.w;
  }
  xn[i] = 0.5f * sx;
  yn[i] = 0.5f * sy;
  f[i] = 0.f;
  g[i] = 0.f;
}

// ---- S = X * Y^T via V_WMMA_F32_16X16X4_F32 (one wave = one 16x16 tile) --
__global__ void gemm_s(const float* __restrict__ X, const float* __restrict__ Y,
                       float* __restrict__ S) {
  const int lane = threadIdx.x;                    // 0..31, one wave32
  const int mn   = lane & 15;
  const int kk   = (lane >> 4) * 2;                // 0 or 2
  const int ib   = blockIdx.y * 16;
  const int jb   = blockIdx.x * 16;
  const float* xrow = X + (size_t)(ib + mn) * D;   // A lane: M = lane&15
  const float* yrow = Y + (size_t)(jb + mn) * D;   // B lane: N = lane&15 (B = Y^T)
  v8f c = {};
#pragma unroll
  for (int k = 0; k < D; k += 4) {
    v2f a = *(const v2f*)(xrow + k + kk);          // K = k + kk + {0,1}
    v2f b = *(const v2f*)(yrow + k + kk);
    c = __builtin_amdgcn_wmma_f32_16x16x4_f32(false, a, false, b,
                                              (short)0, c, false, false);
  }
  // C layout: VGPR r -> M = r (lanes 0-15) / M = r+8 (lanes 16-31), N = lane&15
  const int row0 = ib + (lane >> 4) * 8;
  float* out = S + (size_t)row0 * N + jb + mn;
#pragma unroll
  for (int r = 0; r < 8; ++r) out[(size_t)r * N] = c[r];
}

// ---- f update: one block per row; blocked max-then-sum LSE ---------------
template <bool NT>
__global__ void f_update(const float* __restrict__ S, const float* __restrict__ g,
                         const float* __restrict__ xn, const float* __restrict__ yn,
                         float* __restrict__ f, int row0) {
  const int i = row0 + blockIdx.x;
  const int t = threadIdx.x;                       // 256 threads
  const float xi = xn[i];
  const v4f* row = (const v4f*)(S + (size_t)i * N);
  const v4f* g4  = (const v4f*)g;
  const v4f* yn4 = (const v4f*)yn;
  float v[32];                                     // 8192/256 values per thread
#pragma unroll
  for (int kk = 0; kk < 8; ++kk) {
    int idx = t + kk * 256;
    v4f sv = ld4<NT>(row + idx);
    v4f gv = g4[idx];
    v4f yv = yn4[idx];
    v[4 * kk + 0] = (gv.x - fmaxf(xi + yv.x - sv.x, 0.f)) * INV_EPS;
    v[4 * kk + 1] = (gv.y - fmaxf(xi + yv.y - sv.y, 0.f)) * INV_EPS;
    v[4 * kk + 2] = (gv.z - fmaxf(xi + yv.z - sv.z, 0.f)) * INV_EPS;
    v[4 * kk + 3] = (gv.w - fmaxf(xi + yv.w - sv.w, 0.f)) * INV_EPS;
  }
  float m = v[0];
#pragma unroll
  for (int k = 1; k < 32; ++k) m = fmaxf(m, v[k]);
  float s = 0.f;
#pragma unroll
  for (int k = 0; k < 32; ++k) s += __expf(v[k] - m);

  __shared__ float sm[256], ss[256];
  sm[t] = m; ss[t] = s;
  __syncthreads();
  for (int off = 128; off >= 1; off >>= 1) {
    if (t < off) {
      float mm = sm[t], sv = ss[t];
      lse_merge(mm, sv, sm[t + off], ss[t + off]);
      sm[t] = mm; ss[t] = sv;
    }
    __syncthreads();
  }
  if (t == 0)
    f[i] = TAU * (EPS * LOGW - EPS * (sm[0] + __logf(ss[0])));
}

// ---- g update pass 1: partial column LSE; 8-row blocked exp --------------
template <bool NT>
__global__ void g_partial(const float* __restrict__ S, const float* __restrict__ fp,
                          const float* __restrict__ xn, const float* __restrict__ yn,
                          float* __restrict__ pm, float* __restrict__ ps, int c0) {
  const int j  = blockIdx.x * 256 + threadIdx.x;   // column
  const int c  = c0 + blockIdx.y;                  // row chunk
  const int i0 = c * CHROWS;
  const float ynj = yn[j];
  float m = -__builtin_inff(), s = 0.f;
  for (int i = i0; i < i0 + CHROWS; i += 8) {
    float v[8];
#pragma unroll
    for (int u = 0; u < 8; ++u) {
      float sij = ld1<NT>(S + (size_t)(i + u) * N + j);   // coalesced b32
      v[u] = (fp[i + u] - fmaxf(xn[i + u] + ynj - sij, 0.f)) * INV_EPS;
    }
    float mc = v[0];
#pragma unroll
    for (int u = 1; u < 8; ++u) mc = fmaxf(mc, v[u]);
    float sc = 0.f;
#pragma unroll
    for (int u = 0; u < 8; ++u) sc += __expf(v[u] - mc);
    lse_merge(m, s, mc, sc);
  }
  pm[(size_t)c * N + j] = m;
  ps[(size_t)c * N + j] = s;
}

// ---- g update pass 2: merge chunk partials -------------------------------
__global__ void g_combine(const float* __restrict__ pm, const float* __restrict__ ps,
                          float* __restrict__ g) {
  const int j = blockIdx.x * 256 + threadIdx.x;
  float m = -__builtin_inff(), s = 0.f;
#pragma unroll 4
  for (int c = 0; c < RCH; ++c)
    lse_merge(m, s, pm[(size_t)c * N + j], ps[(size_t)c * N + j]);
  g[j] = TAU * (EPS * LOGW - EPS * (m + __logf(s)));
}

// ---- total_mass partials over 256x256 tiles ------------------------------
template <bool NT>
__global__ void mass_partial(const float* __restrict__ S, const float* __restrict__ f,
                             const float* __restrict__ g, const float* __restrict__ xn,
                             const float* __restrict__ yn, float* __restrict__ part,
                             int c0) {
  const int t  = threadIdx.x;
  const int j  = blockIdx.x * 256 + t;
  const int c  = c0 + blockIdx.y;
  const int i0 = c * CHROWS;
  const float ynj = yn[j], gj = g[j];
  float acc = 0.f;
  for (int i = i0; i < i0 + CHROWS; ++i) {
    float sij = ld1<NT>(S + (size_t)i * N + j);
    float cij = fmaxf(xn[i] + ynj - sij, 0.f);
    acc += __expf((f[i] + gj - cij) * INV_EPS);
  }
  __shared__ float red[256];
  red[t] = acc;
  __syncthreads();
  for (int off = 128; off >= 1; off >>= 1) {
    if (t < off) red[t] += red[t + off];
    __syncthreads();
  }
  if (t == 0) part[c * RCH + blockIdx.x] = red[0];
}

// ---- final: KL-Legendre divergence terms + mass --------------------------
__global__ void finalize(const float* __restrict__ f, const float* __restrict__ g,
                         const float* __restrict__ part, float* __restrict__ out) {
  const int t = threadIdx.x;
  float msum = 0.f;
  for (int k = t; k < RCH * RCH; k += 256) msum += part[k];
  float da = 0.f, db = 0.f;
  for (int i = t; i < N; i += 256) {
    da += __expf(-(f[i] - POT_W) * (1.0f / RHO)) - 1.0f;
    db += __expf(-(g[i] - POT_W) * (1.0f / RHO)) - 1.0f;
  }
  __shared__ float r0[256], r1[256], r2[256];
  r0[t] = msum; r1[t] = da; r2[t] = db;
  __syncthreads();
  for (int off = 128; off >= 1; off >>= 1) {
    if (t < off) { r0[t] += r0[t + off]; r1[t] += r1[t + off]; r2[t] += r2[t + off]; }
    __syncthreads();
  }
  if (t == 0) {
    float div_a = -(1.0f / N) * RHO * r1[0];
    float div_b = -(1.0f / N) * RHO * r2[0];
    out[0] = div_a + div_b + EPS * (1.0f - r0[0]);
  }
}

extern "C" void kernel_launch(void* const* d_in, const int* in_sizes, int n_in,
                              void* d_out, int out_size, void* d_ws, size_t ws_size,
                              hipStream_t stream) {
  (void)in_sizes; (void)n_in; (void)out_size; (void)ws_size;
  const float* X = (const float*)d_in[0];
  const float* Y = (const float*)d_in[1];
  float* out = (float*)d_out;

  // workspace layout (floats): S | xn | yn | f | g | pm | ps | part
  float* S    = (float*)d_ws;
  float* xn   = S + (size_t)N * N;
  float* yn   = xn + N;
  float* f    = yn + N;
  float* g    = f + N;
  float* pm   = g + N;
  float* ps   = pm + (size_t)RCH * N;
  float* part = ps + (size_t)RCH * N;

  setup_kernel<<<N / 256, 256, 0, stream>>>(X, Y, xn, yn, f, g);
  gemm_s<<<dim3(N / 16, N / 16), 32, 0, stream>>>(X, Y, S);

  for (int it = 0; it < ITERS; ++it) {
    // f update: resident rows (RT hits in L2), then streamed rows (NT)
    f_update<false><<<RES_ROWS, 256, 0, stream>>>(S, g, xn, yn, f, 0);
    f_update<true><<<N - RES_ROWS, 256, 0, stream>>>(S, g, xn, yn, f, RES_ROWS);
    // g update: partials over resident / streamed chunks, then merge
    g_partial<false><<<dim3(N / 256, RES_CH), 256, 0, stream>>>(S, f, xn, yn, pm, ps, 0);
    g_partial<true><<<dim3(N / 256, RCH - RES_CH), 256, 0, stream>>>(S, f, xn, yn, pm, ps, RES_CH);
    g_combine<<<N / 256, 256, 0, stream>>>(pm, ps, g);
  }

  mass_partial<false><<<dim3(N / 256, RES_CH), 256, 0, stream>>>(S, f, g, xn, yn, part, 0);
  mass_partial<true><<<dim3(N / 256, RCH - RES_CH), 256, 0, stream>>>(S, f, g, xn, yn, part, RES_CH);
  finalize<<<1, 256, 0, stream>>>(f, g, part, out);
}